// GRACE_21775484191264
// MI455X (gfx1250) — compile-verified
//
#include <hip/hip_runtime.h>
#include <hip/hip_bf16.h>

typedef __attribute__((ext_vector_type(2))) float v2f;
typedef __attribute__((ext_vector_type(8))) float v8f;

// ---------------------------------------------------------------------------
// fp32 WMMA GEMM:  H[N,COUT] = X[N,128] @ W[128,COUT]
// 8 waves / block, W staged in LDS in a K-paired layout so each B fragment is
// a single ds_load_b64 into an even-aligned VGPR pair. Each wave owns a
// 16-row tile and all COUT/16 column tiles; K consumed 4 at a time via
// V_WMMA_F32_16X16X4_F32 (exact fp32 — GEMM is not the bottleneck, precision
// is free).
// ---------------------------------------------------------------------------
template <int COUT>
__global__ __launch_bounds__(256) void gcn_gemm_wmma(
    const float* __restrict__ X, const float* __restrict__ W,
    float* __restrict__ H, int rowTiles) {
  constexpr int K = 128;
  constexpr int NT = COUT / 16;
  // paired layout: sW[p * COUT + n] = { W[2p][n], W[2p+1][n] }
  __shared__ v2f sW[(K / 2) * COUT];

  const int tid = threadIdx.x;
  for (int i = tid; i < (K / 2) * COUT; i += 256) {
    const int p = i / COUT;
    const int n = i - p * COUT;
    v2f w;
    w.x = W[(2 * p) * COUT + n];
    w.y = W[(2 * p + 1) * COUT + n];
    sW[i] = w;
  }
  __syncthreads();

  const int wave = tid >> 5;
  const int lane = tid & 31;
  const int tileM = blockIdx.x * 8 + wave;
  if (tileM >= rowTiles) return;

  const int m0    = tileM * 16;
  const int lrow  = m0 + (lane & 15);  // A: row per lane (N is /16)
  const int khalf = (lane >> 4) * 2;   // A/B: K pair base for this lane half
  const int lcol  = lane & 15;         // B/D: column per lane

  v8f acc[NT] = {};

  const float* xrow = X + (long)lrow * K + khalf;
#pragma unroll 4
  for (int k0 = 0; k0 < K; k0 += 4) {
    v2f a = *(const v2f*)(xrow + k0);  // A frag: 2 consecutive K values
    // B frag: pair index p = (k0 + khalf)/2 = k0/2 + (lane>>4)
    const v2f* bp = sW + ((k0 >> 1) + (lane >> 4)) * COUT + lcol;
#pragma unroll
    for (int t = 0; t < NT; ++t) {
      v2f b = bp[t * 16];              // single ds_load_b64, natural pair
      acc[t] = __builtin_amdgcn_wmma_f32_16x16x4_f32(
          /*neg_a=*/false, a, /*neg_b=*/false, b,
          /*c_mod=*/(short)0, acc[t], /*reuse_a=*/false, /*reuse_b=*/false);
    }
  }

  // D layout: VGPR r, lane l -> M = r + 8*(l>>4), N = l&15
  float* hrow = H + (long)(m0 + 8 * (lane >> 4)) * COUT + lcol;
#pragma unroll
  for (int t = 0; t < NT; ++t) {
#pragma unroll
    for (int r = 0; r < 8; ++r) {
      hrow[(long)r * COUT + t * 16] = acc[t][r];
    }
  }
}

// ---------------------------------------------------------------------------
// Degree / normalization kernels
// ---------------------------------------------------------------------------
__global__ void deg_init(float* deg, int N) {
  int i = blockIdx.x * blockDim.x + threadIdx.x;
  if (i < N) deg[i] = 1.0f;  // self-loop fill value
}

__global__ void deg_acc(const int* __restrict__ dst, float* deg, int E) {
  int e = blockIdx.x * blockDim.x + threadIdx.x;
  if (e < E) unsafeAtomicAdd(&deg[dst[e]], 1.0f);
}

__global__ void deg_rsqrt(float* deg, int N) {
  int i = blockIdx.x * blockDim.x + threadIdx.x;
  if (i < N) deg[i] = rsqrtf(deg[i]);  // deg buffer now holds d_inv
}

// ---------------------------------------------------------------------------
// agg[i,:] = h[i,:] * d_inv[i]^2   (self-loop term; also initializes agg)
// ---------------------------------------------------------------------------
template <int COUT>
__global__ void self_init(const float* __restrict__ h,
                          const float* __restrict__ dinv,
                          float* __restrict__ agg, int N) {
  int idx = blockIdx.x * blockDim.x + threadIdx.x;
  if (idx < N * COUT) {
    int i = idx / COUT;
    float di = dinv[i];
    agg[idx] = h[idx] * (di * di);
  }
}

// ---------------------------------------------------------------------------
// One wave32 per edge: lanes cooperatively move one COUT-channel row.
// Lane loads COUT/32 contiguous floats of h[src] (one coalesced b128/b64 per
// lane; 32 lanes cover the whole row) and scatters with native
// global_atomic_add_f32. h (51 MB) is L2-resident, so this phase runs at
// L2/atomic throughput, not HBM.
// ---------------------------------------------------------------------------
template <int COUT>
__global__ void edge_scatter(const int* __restrict__ src,
                             const int* __restrict__ dst,
                             const float* __restrict__ dinv,
                             const float* __restrict__ h,
                             float* __restrict__ agg, int E) {
  constexpr int V = COUT / 32;
  int t = blockIdx.x * blockDim.x + threadIdx.x;
  int e = t >> 5;
  int lane = t & 31;
  if (e >= E) return;
  int s = src[e];
  int d = dst[e];
  float norm = dinv[s] * dinv[d];
  const float* hp = h + (long)s * COUT + lane * V;
  float* ap = agg + (long)d * COUT + lane * V;
  float vals[V];
#pragma unroll
  for (int j = 0; j < V; ++j) vals[j] = hp[j];  // merges into b128/b64 load
#pragma unroll
  for (int j = 0; j < V; ++j) unsafeAtomicAdd(ap + j, vals[j] * norm);
}

// ---------------------------------------------------------------------------
// out = relu(agg + bias)  (may run in place)
// ---------------------------------------------------------------------------
template <int COUT>
__global__ void bias_relu(const float* agg, const float* __restrict__ b,
                          float* out, int N) {
  int idx = blockIdx.x * blockDim.x + threadIdx.x;
  if (idx < N * COUT) {
    float v = agg[idx] + b[idx & (COUT - 1)];
    out[idx] = fmaxf(v, 0.0f);
  }
}

// ---------------------------------------------------------------------------
extern "C" void kernel_launch(void* const* d_in, const int* in_sizes, int n_in,
                              void* d_out, int out_size, void* d_ws,
                              size_t ws_size, hipStream_t stream) {
  const float* x  = (const float*)d_in[0];
  const int*   ei = (const int*)d_in[1];
  const float* W1 = (const float*)d_in[2];
  const float* b1 = (const float*)d_in[3];
  const float* W2 = (const float*)d_in[4];
  const float* b2 = (const float*)d_in[5];

  const int N = in_sizes[0] / 128;
  const int E = in_sizes[1] / 2;
  const int* src = ei;        // edge_index[0]
  const int* dst = ei + E;    // edge_index[1]
  float* out = (float*)d_out; // [N,64]

  // workspace carve-up (~103 MB)
  float* h0  = (float*)d_ws;           // [N,128] : x@W1, later h1@W2 ([N,64])
  float* agg = h0 + (size_t)N * 128;   // [N,128] : layer-1 agg -> h1
  float* deg = agg + (size_t)N * 128;  // [N]     : degree -> d_inv

  const int thr = 256;
  const int rowTiles = (N + 15) / 16;
  const int gemmBlocks = (rowTiles + 7) / 8;

  // --- normalization (shared by both layers)
  deg_init<<<(N + thr - 1) / thr, thr, 0, stream>>>(deg, N);
  deg_acc<<<(E + thr - 1) / thr, thr, 0, stream>>>(dst, deg, E);
  deg_rsqrt<<<(N + thr - 1) / thr, thr, 0, stream>>>(deg, N);

  // --- layer 1: h0 = x @ W1 ; agg = scatter(norm * h0[src]) + self ; relu+b1
  gcn_gemm_wmma<128><<<gemmBlocks, thr, 0, stream>>>(x, W1, h0, rowTiles);
  self_init<128><<<(N * 128 + thr - 1) / thr, thr, 0, stream>>>(h0, deg, agg, N);
  edge_scatter<128><<<(E * 32 + thr - 1) / thr, thr, 0, stream>>>(src, dst, deg,
                                                                  h0, agg, E);
  bias_relu<128><<<(N * 128 + thr - 1) / thr, thr, 0, stream>>>(agg, b1, agg, N);

  // --- layer 2: h0 = h1 @ W2 ; out = scatter + self ; relu+b2
  gcn_gemm_wmma<64><<<gemmBlocks, thr, 0, stream>>>(agg, W2, h0, rowTiles);
  self_init<64><<<(N * 64 + thr - 1) / thr, thr, 0, stream>>>(h0, deg, out, N);
  edge_scatter<64><<<(E * 32 + thr - 1) / thr, thr, 0, stream>>>(src, dst, deg,
                                                                 h0, out, E);
  bias_relu<64><<<(N * 64 + thr - 1) / thr, thr, 0, stream>>>(out, b2, out, N);
}